// GaussianMomentDescriptor_35338990911689
// MI455X (gfx1250) — compile-verified
//
#include <hip/hip_runtime.h>
#include <math.h>

#define N_ATOMS   20000
#define N_NEIGH   800000
#define N_RADIAL  5
#define N_BASIS   7
#define N_SPECIES 10
#define R_MAX     6.0f
#define R_MIN     0.5f
#define PI_F      3.14159265358979f
#define MOM_PER_ATOM 200   // layout: [r][40], c: 0=m0, 1..3=m1_i, 4..12=m2_(3i+j), 13..39=m3_(3(3i+j)+k)
#define OUT_PER_ATOM 360
#define WPB 8              // waves per block in contraction kernel (wave32)

typedef __attribute__((ext_vector_type(2))) float v2f;
typedef __attribute__((ext_vector_type(8))) float v8f;

// lower-triangle index tables (match np.tril_indices ordering in the reference)
__constant__ unsigned char T2I[15] = {0,1,1,2,2,2,3,3,3,3,4,4,4,4,4};
__constant__ unsigned char T2J[15] = {0,0,1,0,1,2,0,1,2,3,0,1,2,3,4};
__constant__ unsigned char T3I[35] = {0, 1,1,1, 2,2,2,2,2,2, 3,3,3,3,3,3,3,3,3,3, 4,4,4,4,4,4,4,4,4,4,4,4,4,4,4};
__constant__ unsigned char T3J[35] = {0, 0,1,1, 0,1,1,2,2,2, 0,1,1,2,2,2,3,3,3,3, 0,1,1,2,2,2,3,3,3,3,4,4,4,4,4};
__constant__ unsigned char T3K[35] = {0, 0,0,1, 0,0,1,0,1,2, 0,0,1,0,1,2,0,1,2,3, 0,0,1,0,1,2,0,1,2,3,0,1,2,3,4};

// ---------------- kernel 0: zero the moment accumulator ----------------
__global__ void gm_zero(float* __restrict__ M, int n) {
  int i = blockIdx.x * blockDim.x + threadIdx.x;
  int stride = gridDim.x * blockDim.x;
  for (; i < n; i += stride) M[i] = 0.0f;
}

// ---------------- kernel 1: per-neighbor radial + moment scatter ----------------
__global__ __launch_bounds__(256)
void gm_scatter(const float* __restrict__ dr_vec, const int* __restrict__ Z,
                const int* __restrict__ nbr, const float* __restrict__ emb,
                float* __restrict__ M) {
  int n = blockIdx.x * blockDim.x + threadIdx.x;
  if (n >= N_NEIGH) return;
  int ii = nbr[n];
  int jj = nbr[N_NEIGH + n];
  if (ii == jj) return;                       // (idx_i != idx_j) mask -> zero contribution
  float x = dr_vec[3*n + 0];
  float y = dr_vec[3*n + 1];
  float z = dr_vec[3*n + 2];
  float dr = sqrtf(x*x + y*y + z*z);
  if (!(dr < R_MAX)) return;                  // cutoff factor is zero -> zero contribution
  float inv = 1.0f / (dr + 1e-5f);
  float dn0 = x * inv, dn1 = y * inv, dn2 = z * inv;
  float cut = 0.5f * (cosf(PI_F * dr * (1.0f / R_MAX)) + 1.0f);

  const float betta = (float)(N_BASIS * N_BASIS) / (R_MAX * R_MAX);
  const float rad_norm = powf(2.0f * betta / PI_F, 0.75f);   // constant-folded
  float basis[N_BASIS];
#pragma unroll
  for (int b = 0; b < N_BASIS; ++b) {
    float d = dr - (R_MIN + ((R_MAX - R_MIN) / (float)N_BASIS) * (float)b);
    basis[b] = rad_norm * expf(-betta * d * d);
  }

  int zi = Z[ii], zj = Z[jj];
  const float* co = emb + (size_t)(zi * N_SPECIES + zj) * (N_RADIAL * N_BASIS);
  const float scale = cut * 0.3779644730092272f;             // 1/sqrt(N_BASIS)
  float radial[N_RADIAL];
#pragma unroll
  for (int r = 0; r < N_RADIAL; ++r) {
    float s = 0.0f;
#pragma unroll
    for (int b = 0; b < N_BASIS; ++b) s += co[r * N_BASIS + b] * basis[b];
    radial[r] = s * scale;
  }

  // P = [1, dn_i, dn_i dn_j, dn_i dn_j dn_k] (40 entries); moments = radial[r] * P[c]
  float P[40];
  P[0] = 1.0f; P[1] = dn0; P[2] = dn1; P[3] = dn2;
#pragma unroll
  for (int i = 0; i < 3; ++i)
#pragma unroll
    for (int j = 0; j < 3; ++j) P[4 + 3*i + j] = P[1 + i] * P[1 + j];
#pragma unroll
  for (int u = 0; u < 9; ++u)
#pragma unroll
    for (int k = 0; k < 3; ++k) P[13 + 3*u + k] = P[4 + u] * P[1 + k];

  float* Ma = M + (size_t)jj * MOM_PER_ATOM;   // segment_sum over idx_j
#pragma unroll
  for (int r = 0; r < N_RADIAL; ++r) {
    float rv = radial[r];
#pragma unroll
    for (int c = 0; c < 40; ++c) {
      __hip_atomic_fetch_add(&Ma[r * 40 + c], rv * P[c],
                             __ATOMIC_RELAXED, __HIP_MEMORY_SCOPE_AGENT);
    }
  }
}

// ---------------- kernel 2: per-atom contractions (one wave32 per atom) ----------------
// G = X * X^T with X[(3r+k), (3i+j)] = m3[r,i,j,k]  (15x9, padded to 16x12)
// -> D[r,s,k,l] = G[3r+k, 3s+l]  (drives contr_6), block-trace drives contr_3.
__global__ __launch_bounds__(256)
void gm_contract(const float* __restrict__ M, float* __restrict__ out) {
  __shared__ float s_mom[WPB][MOM_PER_ATOM];
  __shared__ float s_G[WPB][256];
  const int lane = threadIdx.x & 31;
  const int w    = threadIdx.x >> 5;
  const int a    = blockIdx.x * WPB + w;
  const bool active = (a < N_ATOMS);
  float* sm = s_mom[w];
  float* sg = s_G[w];

  if (active) {
    const float* Ma = M + (size_t)a * MOM_PER_ATOM;
    for (int c = lane; c < MOM_PER_ATOM; c += 32) sm[c] = Ma[c];
  }
  __syncthreads();

  if (active) {
    const int m  = lane & 15;            // output row index this lane feeds (A layout)
    const int r  = m / 3;
    const int k  = m - 3 * r;
    const bool hi = (lane >= 16);
    const bool mOK = (m < 15);
#if __has_builtin(__builtin_amdgcn_wmma_f32_16x16x4_f32)
    v8f acc = {};
#pragma unroll
    for (int kk = 0; kk < 3; ++kk) {
      // A 16x4 f32 layout: vgpr0 = K(+0) lanes0-15 / K(+2) lanes16-31; vgpr1 = K(+1)/K(+3)
      // B 4x16 layout mirrors it, and B = X^T, so B's VGPRs equal A's.
      int K0 = 4 * kk + (hi ? 2 : 0);
      int K1 = K0 + 1;
      float a0 = (mOK && K0 < 9) ? sm[r * 40 + 13 + 3 * K0 + k] : 0.0f;
      float a1 = (mOK && K1 < 9) ? sm[r * 40 + 13 + 3 * K1 + k] : 0.0f;
      v2f A = {a0, a1};
      acc = __builtin_amdgcn_wmma_f32_16x16x4_f32(false, A, false, A,
                                                  (short)0, acc, false, false);
    }
    // C/D layout: VGPR v, lane l -> row = v + 8*(l>=16), col = l&15
    const int rbase = hi ? 8 : 0;
#pragma unroll
    for (int v = 0; v < 8; ++v) sg[(v + rbase) * 16 + m] = acc[v];
#else
    // scalar fallback (keeps compile green if the f32 WMMA builtin is absent)
    for (int idx = lane; idx < 15 * 15; idx += 32) {
      int mm = idx / 15, nn = idx % 15;
      int rm = mm / 3, km = mm - 3 * rm, rn = nn / 3, kn = nn - 3 * rn;
      float s = 0.0f;
      for (int u = 0; u < 9; ++u)
        s += sm[rm * 40 + 13 + 3 * u + km] * sm[rn * 40 + 13 + 3 * u + kn];
      sg[mm * 16 + nn] = s;
    }
#endif
  }
  __syncthreads();

  if (active) {
    float* dst = out + (size_t)a * OUT_PER_ATOM;
    for (int o = lane; o < OUT_PER_ATOM; o += 32) {
      float val = 0.0f;
      if (o < 5) {                                   // contr_0: m0
        val = sm[o * 40];
      } else if (o < 20) {                           // contr_1: m1.m1 (tril2)
        int p = o - 5; int r = T2I[p], s = T2J[p];
        for (int i = 0; i < 3; ++i) val += sm[r*40+1+i] * sm[s*40+1+i];
      } else if (o < 35) {                           // contr_2: m2.m2 (tril2)
        int p = o - 20; int r = T2I[p], s = T2J[p];
        for (int u = 0; u < 9; ++u) val += sm[r*40+4+u] * sm[s*40+4+u];
      } else if (o < 50) {                           // contr_3: m3.m3 = block-trace of G
        int p = o - 35; int r = T2I[p], s = T2J[p];
        for (int k = 0; k < 3; ++k) val += sg[(3*r+k)*16 + 3*s+k];
      } else if (o < 85) {                           // contr_4: m2 m2 m2 (tril3)
        int q = o - 50; int r = T3I[q], s = T3J[q], t = T3K[q];
        for (int i = 0; i < 3; ++i)
          for (int j = 0; j < 3; ++j)
            for (int k = 0; k < 3; ++k)
              val += sm[r*40+4+3*i+j] * sm[s*40+4+3*i+k] * sm[t*40+4+3*j+k];
      } else if (o < 160) {                          // contr_5: m1 m1 m2 (tril2 x 5)
        int e = o - 85; int p = e / 5, t = e - 5 * p;
        int r = T2I[p], s = T2J[p];
        for (int i = 0; i < 3; ++i)
          for (int j = 0; j < 3; ++j)
            val += sm[r*40+1+i] * sm[s*40+1+j] * sm[t*40+4+3*i+j];
      } else if (o < 235) {                          // contr_6: D[r,s,k,l]*m2[t,k,l] via WMMA G
        int e = o - 160; int p = e / 5, t = e - 5 * p;
        int r = T2I[p], s = T2J[p];
        for (int k = 0; k < 3; ++k)
          for (int l = 0; l < 3; ++l)
            val += sg[(3*r+k)*16 + 3*s+l] * sm[t*40+4+3*k+l];
      } else {                                       // contr_7: m3 m2 m1 (full 125)
        int e = o - 235; int r = e / 25, s = (e / 5) % 5, t = e % 5;
        for (int i = 0; i < 3; ++i)
          for (int j = 0; j < 3; ++j)
            for (int k = 0; k < 3; ++k)
              val += sm[r*40+13+3*(3*i+j)+k] * sm[s*40+4+3*i+j] * sm[t*40+1+k];
      }
      dst[o] = val;
    }
  }
}

extern "C" void kernel_launch(void* const* d_in, const int* in_sizes, int n_in,
                              void* d_out, int out_size, void* d_ws, size_t ws_size,
                              hipStream_t stream) {
  const float* dr_vec = (const float*)d_in[0];   // (800000, 3) f32
  const int*   Z      = (const int*)d_in[1];     // (20000,)    i32
  const int*   nbr    = (const int*)d_in[2];     // (2, 800000) i32
  const float* emb    = (const float*)d_in[3];   // (10,10,5,7) f32
  float* out = (float*)d_out;                    // (20000, 360) f32
  float* M   = (float*)d_ws;                     // 20000*200*4 = 16 MB accumulator (L2-resident)

  const int momTotal = N_ATOMS * MOM_PER_ATOM;
  gm_zero<<<(momTotal + 1023) / 1024, 256, 0, stream>>>(M, momTotal);
  gm_scatter<<<(N_NEIGH + 255) / 256, 256, 0, stream>>>(dr_vec, Z, nbr, emb, M);
  gm_contract<<<(N_ATOMS + WPB - 1) / WPB, 256, 0, stream>>>(M, out);
}